// KANLayer_72035191488889
// MI455X (gfx1250) — compile-verified
//
#include <hip/hip_runtime.h>
#include <hip/hip_bf16.h>
#include <stdint.h>

#define IN_DIM   256
#define OUT_DIM  256
#define GPTS     12     // NUM+1+2K knots
#define NK       8      // NUM+K coefficients
#define BATCH    2048

#define B_TILE   64     // batch rows per workgroup (4 waves x 16)
#define O_TILE   16     // output cols per workgroup
#define I_CHUNK  8      // input dims staged per LDS round
#define NTHREADS 128
#define N_CHUNKS (IN_DIM / I_CHUNK)
#define FLUSH_IT ((B_TILE * O_TILE * 2) / NTHREADS)   // 16

// LDS strides (floats): padded so flush quads are 16B aligned and writes avoid bank conflicts
#define ISTRIDE  12                  // 8 data + 4 pad
#define BSTRIDE  (O_TILE * ISTRIDE)  // 192

typedef float v2f __attribute__((ext_vector_type(2)));
typedef float v8f __attribute__((ext_vector_type(8)));
typedef unsigned int v4u __attribute__((ext_vector_type(4)));
typedef int v8i __attribute__((ext_vector_type(8)));
typedef int v4i __attribute__((ext_vector_type(4)));

#if defined(__HIP_DEVICE_COMPILE__) && defined(__gfx1250__)
#if __has_builtin(__builtin_amdgcn_wmma_f32_16x16x4_f32)
#define HAS_WMMA_F32 1
#endif
#if __has_builtin(__builtin_amdgcn_tensor_load_to_lds) && __has_builtin(__builtin_amdgcn_s_wait_tensorcnt)
#define HAS_TDM 1
#endif
#endif

// Fast reciprocal: v_rcp_f32 (TRANS pipe, co-executes with VALU on CDNA5)
__device__ __forceinline__ float fast_rcp(float v) {
#if defined(__HIP_DEVICE_COMPILE__) && __has_builtin(__builtin_amdgcn_rcpf)
  return __builtin_amdgcn_rcpf(v);
#else
  return 1.f / v;
#endif
}

#if defined(HAS_TDM)
// Issue one 2D TDM transfer: coef[i0..i0+7][o0..o0+15][0..7] -> dst (contiguous 4KB).
__device__ __forceinline__ void tdm_issue_coef(const float* coefp, int o0, int i0,
                                               float* dst) {
  const float* gsrc = coefp + (size_t)i0 * (OUT_DIM * NK) + (size_t)o0 * NK;
  uint64_t ga  = (uint64_t)(uintptr_t)gsrc;
  uint32_t lds = (uint32_t)(uintptr_t)dst;
  // D# group0: count=1 | lds_addr | global_addr[56:0] | type=2
  v4u g0 = { 1u, lds, (uint32_t)ga,
             (uint32_t)((ga >> 32) & 0x1FFFFFFu) | 0x80000000u };
  // D# group1: data_size=4B; tensor_dim0=128, tensor_dim1=8;
  //            tile_dim0=128 (o*k row), tile_dim1=8 (i), dim0_stride=2048 floats
  v8i g1 = { 0x20000, (int)(128u << 16), (int)(8u << 16), (int)(128u << 16),
             8, 2048, 0, 0 };
  v4i gz4 = { 0, 0, 0, 0 };
  v8i gz8 = { 0, 0, 0, 0, 0, 0, 0, 0 };
  __builtin_amdgcn_tensor_load_to_lds(g0, g1, gz4, gz4, gz8, 0);
}
#endif

__global__ __launch_bounds__(NTHREADS)
void kan_main(const float* __restrict__ xp, const float* __restrict__ gridp,
              const float* __restrict__ coefp, const float* __restrict__ sbp,
              const float* __restrict__ sspp, const float* __restrict__ mp,
              float* __restrict__ o_ysum, float* __restrict__ o_post,
              float* __restrict__ o_spl)
{
#if defined(__HIP_DEVICE_COMPILE__)
  __shared__ float yspLDS[B_TILE * BSTRIDE];           // 49152 B: spline results [b][o][i]
  __shared__ float coefLDS[2][I_CHUNK * O_TILE * NK];  //  8192 B: double-buffered coef tiles
  __shared__ float tabLDS[3 * O_TILE * I_CHUNK];       //  1536 B: sb/ssp/mask transposed [o][i]
  __shared__ float siluLDS[B_TILE * I_CHUNK];          //  2048 B: silu(x) [b][i]

  const int tid  = threadIdx.x;
  const int w    = tid >> 5;     // wave id (4 waves)
  const int lane = tid & 31;
  const int n    = lane & 15;    // N index (output col) / batch row within wave tile
  const int h    = lane >> 4;    // half-wave select (K pairs / M+8)

  const int bBlk = blockIdx.x >> 4;
  const int oBlk = blockIdx.x & 15;
  const int B0   = bBlk * B_TILE;
  const int o0   = oBlk * O_TILE;

  const int bL0  = w * 16;
  const size_t bRow = (size_t)(B0 + bL0 + n);  // batch row whose bsp this lane computes

  // Per-thread y_sum partials: flush task (it,tid) owns a fixed (b,o,half) forever.
  float ysr[FLUSH_IT];
  #pragma unroll
  for (int it = 0; it < FLUSH_IT; ++it) ysr[it] = 0.f;

#if defined(HAS_TDM)
  if (tid < 32) tdm_issue_coef(coefp, o0, 0, &coefLDS[0][0]);   // prefetch chunk 0
#endif

  for (int chunk = 0; chunk < N_CHUNKS; ++chunk) {
    const int i0  = chunk * I_CHUNK;
    const int buf = chunk & 1;

    // ---- coef staging: TDM double-buffer prefetch, else cooperative loads ----
#if defined(HAS_TDM)
    if (tid < 32) {
      if (chunk + 1 < N_CHUNKS) {
        tdm_issue_coef(coefp, o0, (chunk + 1) * I_CHUNK, &coefLDS[(chunk + 1) & 1][0]);
        __builtin_amdgcn_s_wait_tensorcnt(1);   // retire chunk `chunk` (in-order)
      } else {
        __builtin_amdgcn_s_wait_tensorcnt(0);
      }
    }
#else
    for (int e = tid; e < I_CHUNK * O_TILE * NK; e += NTHREADS) {
      int iL = e >> 7, r = e & 127;
      coefLDS[buf][e] = coefp[(size_t)(i0 + iL) * (OUT_DIM * NK) + (size_t)o0 * NK + r];
    }
#endif
    // ---- stage scale_base / scale_sp / mask transposed as [o][i] ----
    for (int e = tid; e < 3 * O_TILE * I_CHUNK; e += NTHREADS) {
      int tsel = e >> 7;
      int r = e & 127;
      int oL = r >> 3, iL = r & 7;
      const float* src = (tsel == 0) ? sbp : ((tsel == 1) ? sspp : mp);
      tabLDS[e] = src[(size_t)(i0 + iL) * OUT_DIM + (o0 + oL)];
    }
    __syncthreads();

    // ---- compute: per i, Cox-de-Boor basis then 16x16x8 GEMM via 2x WMMA f32 ----
    for (int iL = 0; iL < I_CHUNK; ++iL) {
      const int i = i0 + iL;
      float g[GPTS];
      #pragma unroll
      for (int j = 0; j < GPTS; ++j) g[j] = gridp[i * GPTS + j];
      const float x = xp[bRow * IN_DIM + i];
      const float sil = x * fast_rcp(1.f + __expf(-x));   // silu via v_rcp
      if (h == 0) siluLDS[(bL0 + n) * I_CHUNK + iL] = sil;

      float t[11];
      #pragma unroll
      for (int j = 0; j < 11; ++j) t[j] = (x >= g[j] && x < g[j + 1]) ? 1.f : 0.f;
      // Cox-de-Boor: dr(ki,j) == dl(ki,j+1), so one reciprocal array per level,
      // computed just-in-time (<=12 live) on the TRANS pipe instead of IEEE divides.
      #pragma unroll
      for (int ki = 1; ki <= 3; ++ki) {
        float rd[12];
        #pragma unroll
        for (int j = 0; j <= 11 - ki; ++j) {
          float d = g[j + ki] - g[j];
          rd[j] = (d != 0.f) ? fast_rcp(d) : 0.f;
        }
        #pragma unroll
        for (int j = 0; j < 11 - ki; ++j) {
          float left  = (x - g[j]) * rd[j];
          float right = (g[j + ki + 1] - x) * rd[j + 1];
          t[j] = left * t[j] + right * t[j + 1];
        }
      }

      float cres[8];
#if defined(HAS_WMMA_F32)
      // A 16x4 f32: lanes 0-15 hold K=0,1 (v0,v1); lanes 16-31 hold K=2,3
      v2f aLo = { h ? t[2] : t[0], h ? t[3] : t[1] };
      v2f aHi = { h ? t[6] : t[4], h ? t[7] : t[5] };
      const float* cb = &coefLDS[buf][iL * (O_TILE * NK) + n * NK + 2 * h];
      v2f bLo = { cb[0], cb[1] };
      v2f bHi = { cb[4], cb[5] };
      v8f c = { 0.f, 0.f, 0.f, 0.f, 0.f, 0.f, 0.f, 0.f };
      c = __builtin_amdgcn_wmma_f32_16x16x4_f32(false, aLo, false, bLo, (short)0, c, false, false);
      c = __builtin_amdgcn_wmma_f32_16x16x4_f32(false, aHi, false, bHi, (short)0, c, false, false);
      #pragma unroll
      for (int r = 0; r < 8; ++r) cres[r] = c[r];
#else
      const float* ck = &coefLDS[buf][iL * (O_TILE * NK) + n * NK];
      #pragma unroll
      for (int r = 0; r < 8; ++r) {
        int m = r + 8 * h;
        float acc = 0.f;
        #pragma unroll
        for (int k = 0; k < NK; ++k) acc += __shfl(t[k], m, 32) * ck[k];
        cres[r] = acc;
      }
#endif
      // C layout: VGPR r, lane -> (M = r + 8h, N = lane%16)
      #pragma unroll
      for (int r = 0; r < 8; ++r) {
        int bL = bL0 + r + 8 * h;
        yspLDS[bL * BSTRIDE + n * ISTRIDE + iL] = cres[r];
      }
    }
    __syncthreads();

    // ---- flush: i-contiguous float4 stores for postspline & postacts ----
    #pragma unroll
    for (int it = 0; it < FLUSH_IT; ++it) {
      int flat = it * NTHREADS + tid;
      int row  = flat >> 1;      // (b,o) row
      int half = flat & 1;       // which 4-i quad
      int bL = row >> 4;
      int oL = row & 15;
      const float4 ys = *(const float4*)&yspLDS[bL * BSTRIDE + oL * ISTRIDE + half * 4];
      const float4 sb = *(const float4*)&tabLDS[0 * 128 + oL * 8 + half * 4];
      const float4 sp = *(const float4*)&tabLDS[1 * 128 + oL * 8 + half * 4];
      const float4 mk = *(const float4*)&tabLDS[2 * 128 + oL * 8 + half * 4];
      const float4 sl = *(const float4*)&siluLDS[bL * I_CHUNK + half * 4];
      float4 pa;
      pa.x = mk.x * (sb.x * sl.x + sp.x * ys.x);
      pa.y = mk.y * (sb.y * sl.y + sp.y * ys.y);
      pa.z = mk.z * (sb.z * sl.z + sp.z * ys.z);
      pa.w = mk.w * (sb.w * sl.w + sp.w * ys.w);
      ysr[it] += pa.x + pa.y + pa.z + pa.w;
      size_t gq = (((size_t)(B0 + bL) * OUT_DIM) + (size_t)(o0 + oL)) * IN_DIM
                  + (size_t)(i0 + half * 4);
      *(float4*)&o_spl[gq]  = ys;
      *(float4*)&o_post[gq] = pa;
    }
    __syncthreads();
  }

  // ---- y_sum writeout: combine the two half-row partials via shfl_xor ----
  #pragma unroll
  for (int it = 0; it < FLUSH_IT; ++it) {
    float v = ysr[it] + __shfl_xor(ysr[it], 1, 32);
    int flat = it * NTHREADS + tid;
    if ((flat & 1) == 0) {
      int row = flat >> 1;
      int bL = row >> 4, oL = row & 15;
      o_ysum[(size_t)(B0 + bL) * OUT_DIM + (o0 + oL)] = v;
    }
  }
#endif
}

// preacts[b][o][i] = x[b][i] : pure streaming broadcast, fully coalesced
__global__ __launch_bounds__(256)
void kan_preacts(const float* __restrict__ xp, float* __restrict__ o_pre)
{
#if defined(__HIP_DEVICE_COMPILE__)
  size_t q  = (size_t)blockIdx.x * 256 + threadIdx.x; // quad index over [B][OUT][IN/4]
  size_t b  = q >> 14;                                // / (256*64)
  size_t r  = q & 16383;
  size_t iq = r & 63;
  float4 v = *(const float4*)&xp[b * IN_DIM + iq * 4];
  *(float4*)&o_pre[q * 4] = v;
#endif
}

extern "C" void kernel_launch(void* const* d_in, const int* in_sizes, int n_in,
                              void* d_out, int out_size, void* d_ws, size_t ws_size,
                              hipStream_t stream) {
  (void)in_sizes; (void)n_in; (void)out_size; (void)d_ws; (void)ws_size;
  const float* x    = (const float*)d_in[0];
  const float* grid = (const float*)d_in[1];
  const float* coef = (const float*)d_in[2];
  const float* sb   = (const float*)d_in[3];
  const float* ssp  = (const float*)d_in[4];
  const float* mask = (const float*)d_in[5];

  float* out    = (float*)d_out;
  float* o_ysum = out;
  float* o_pre  = o_ysum + (size_t)BATCH * OUT_DIM;
  float* o_post = o_pre  + (size_t)BATCH * OUT_DIM * IN_DIM;
  float* o_spl  = o_post + (size_t)BATCH * OUT_DIM * IN_DIM;

  kan_main<<<dim3((BATCH / B_TILE) * (OUT_DIM / O_TILE)), dim3(NTHREADS), 0, stream>>>(
      x, grid, coef, sb, ssp, mask, o_ysum, o_post, o_spl);
  kan_preacts<<<dim3((unsigned)((size_t)BATCH * OUT_DIM * IN_DIM / 4 / 256)),
                dim3(256), 0, stream>>>(x, o_pre);
}